// ExpNegL2_74655121539837
// MI455X (gfx1250) — compile-verified
//
#include <hip/hip_runtime.h>
#include <math.h>

// ---------------------------------------------------------------------------
// exp(-||normalize(audio) - normalize(text)||) via GEMM on CDNA5 (gfx1250)
//   audio: [B=32, T=4096, D=512] f32,  text: [B=32, M=64, D=512] f32
//   out:   [B, M, T] f32
// Memory-bound (~306 MB HBM => ~13us @ 23.3 TB/s); use f32 WMMA 16x16x4
// so the matrix math stays exact and effectively free.
// ---------------------------------------------------------------------------

typedef __attribute__((ext_vector_type(2))) float v2f;
typedef __attribute__((ext_vector_type(8))) float v8f;

#define BATCHES   32
#define T_ROWS    4096
#define M_ROWS    64
#define DDIM      512
#define TILE_N    32     // audio rows per block
#define EPS_NRM   1e-12f
#define EPS_D2    1e-12f

// ---------------------------------------------------------------------------
// Kernel 1: per-row inverse L2 norm. One wave32 per 512-float row.
// ---------------------------------------------------------------------------
__global__ __launch_bounds__(256) void ExpNegL2_row_invnorm(
    const float* __restrict__ x, float* __restrict__ inv_norm, int nrows) {
  const int row  = blockIdx.x * 8 + (threadIdx.x >> 5);
  const int lane = threadIdx.x & 31;
  if (row >= nrows) return;

  const float4* p = (const float4*)(x + (size_t)row * DDIM);
  float s = 0.f;
#pragma unroll
  for (int i = 0; i < 4; ++i) {
    float4 v = p[lane + 32 * i];
    s += v.x * v.x + v.y * v.y + v.z * v.z + v.w * v.w;
  }
  // wave32 butterfly reduction
#pragma unroll
  for (int off = 16; off > 0; off >>= 1) s += __shfl_xor(s, off, 32);

  if (lane == 0) {
    float n = sqrtf(s);
    inv_norm[row] = 1.0f / fmaxf(n, EPS_NRM);
  }
}

// ---------------------------------------------------------------------------
// Kernel 2: per-batch GEMM  dots[m, t] = sum_d text[m,d]*audio[t,d]
// with fused normalization + exp(-dist) epilogue.
//
// Block: 256 thr = 8 waves. Wave (w): m_tile = (w&3)*16, n_tile = (w>>2)*16.
// Block tile: 64(M) x 32(N). Audio tile staged in LDS, k-packed [k/4][n][4]
// => conflict-free ds_load_b64 B-operand fetches, exactly 64 KB.
// ---------------------------------------------------------------------------
__global__ __launch_bounds__(256) void ExpNegL2_wmma_gemm(
    const float* __restrict__ audio, const float* __restrict__ text,
    const float* __restrict__ inv_na, const float* __restrict__ inv_nt,
    float* __restrict__ out) {
  // LDS: element (row n in 0..31, col k in 0..511) at float index
  //   (k>>2)*128 + n*4 + (k&3)
  __shared__ float lds_a[TILE_N * DDIM];  // 65536 bytes

  const int b  = blockIdx.y;
  const int t0 = blockIdx.x * TILE_N;

  const float* audio_b = audio + ((size_t)b * T_ROWS + t0) * DDIM;
  const float* text_b  = text + (size_t)b * M_ROWS * DDIM;

  // Cooperative stage: 8 threads per audio row, 16 float4 each.
  {
    const int row = threadIdx.x >> 3;   // 0..31
    const int sub = threadIdx.x & 7;    // 0..7
    const float4* src = (const float4*)(audio_b + (size_t)row * DDIM);
#pragma unroll
    for (int i = 0; i < 16; ++i) {
      const int j = sub + 8 * i;              // float4 group index = k/4
      float4 v = src[j];
      *(float4*)(&lds_a[j * (TILE_N * 4) + row * 4]) = v;
    }
  }
  __syncthreads();

  const int wave   = threadIdx.x >> 5;
  const int lane   = threadIdx.x & 31;
  const int m_tile = (wave & 3) << 4;   // 0,16,32,48
  const int n_tile = (wave >> 2) << 4;  // 0,16
  const int half   = lane >> 4;         // 0/1
  const int nn     = lane & 15;

  // A operand: text row (m_tile+nn), float2 at col (k + 2*half)
  const float* a_ptr = text_b + (size_t)(m_tile + nn) * DDIM + 2 * half;
  // B operand: LDS float2 at index 32*k + 4*(n_tile+nn) + 2*half
  const float* b_ptr = &lds_a[4 * (n_tile + nn) + 2 * half];

  v8f acc = {};
#pragma unroll 8
  for (int k = 0; k < DDIM; k += 4) {
    v2f a = *(const v2f*)(a_ptr + k);
    v2f bb = *(const v2f*)(b_ptr + 32 * k);
    acc = __builtin_amdgcn_wmma_f32_16x16x4_f32(
        /*neg_a=*/false, a, /*neg_b=*/false, bb,
        /*c_mod=*/(short)0, acc, /*reuse_a=*/false, /*reuse_b=*/false);
  }

  // Epilogue: D VGPR r -> m = m_tile + r + 8*half, n -> t = t0 + n_tile + nn
  const int t = t0 + n_tile + nn;
  const float ina = inv_na[(size_t)b * T_ROWS + t];
  float* out_b = out + (size_t)b * M_ROWS * T_ROWS;
#pragma unroll
  for (int r = 0; r < 8; ++r) {
    const int m  = m_tile + r + 8 * half;
    const float s    = acc[r] * ina * inv_nt[(size_t)b * M_ROWS + m];
    const float d2   = fmaxf(2.0f - 2.0f * s, 0.0f);
    const float dist = sqrtf(fmaxf(d2, EPS_D2));
    out_b[(size_t)m * T_ROWS + t] = expf(-dist);
  }
}

// ---------------------------------------------------------------------------
extern "C" void kernel_launch(void* const* d_in, const int* in_sizes, int n_in,
                              void* d_out, int out_size, void* d_ws, size_t ws_size,
                              hipStream_t stream) {
  const float* audio = (const float*)d_in[0];
  const float* text  = (const float*)d_in[1];
  float* out = (float*)d_out;

  // Workspace: audio inverse norms (B*T) then text inverse norms (B*M).
  float* inv_na = (float*)d_ws;
  float* inv_nt = inv_na + (size_t)BATCHES * T_ROWS;

  const int audio_rows = BATCHES * T_ROWS;  // 131072
  const int text_rows  = BATCHES * M_ROWS;  // 2048

  ExpNegL2_row_invnorm<<<audio_rows / 8, 256, 0, stream>>>(audio, inv_na, audio_rows);
  ExpNegL2_row_invnorm<<<text_rows / 8, 256, 0, stream>>>(text, inv_nt, text_rows);

  dim3 grid(T_ROWS / TILE_N, BATCHES);  // (128, 32)
  ExpNegL2_wmma_gemm<<<grid, 256, 0, stream>>>(audio, text, inv_na, inv_nt, out);
}